// GTR_50139448214076
// MI455X (gfx1250) — compile-verified
//
#include <hip/hip_runtime.h>
#include <hip/hip_bf16.h>
#include <math.h>

typedef float v2f __attribute__((ext_vector_type(2)));
typedef float v8f __attribute__((ext_vector_type(8)));

#define NNODE_DIM 64
#define GEMM_TB   128          // 4 waves per block
#define GEMM_ROWS 256          // 4 waves x 64 rows
#define MAX_K     256

// ---------------------------------------------------------------------------
// fp32 WMMA GEMM: Out[M,Nc] = A[M,K] @ W[K,Nc] (+bias) (optional relu)
// Block = 4 waves; each wave computes a 64x16 tile (4 accumulators).
// W column-panel (K x 16) staged once per block into LDS as paired-k float2.
// Inner loop software-pipelined; A is addressed as uniform 64-bit base +
// per-lane 32-bit element offsets (all matrices < 4 GB) so the backend can
// use the saddr+voffset global-load form instead of 64-bit VALU adds.
// Tail rows are CLAMPED (D row m depends only on A row m; garbage rows are
// computed but never stored) -- EXEC stays all-ones around WMMA.
// ---------------------------------------------------------------------------
__global__ __launch_bounds__(GEMM_TB)
void gtr_wmma_gemm(const float* __restrict__ A,
                   const float* __restrict__ W,
                   const float* __restrict__ bias,
                   float* __restrict__ Out,
                   int M, int K, int Nc, int relu)
{
    __shared__ float2 wlds[(MAX_K / 2) * 16];   // 16 KB max

    const int tid  = threadIdx.x;
    const int lane = tid & 31;
    const int wave = tid >> 5;
    const int half = lane >> 4;                 // 0: lanes 0-15, 1: lanes 16-31
    const int l16  = lane & 15;
    const int col0 = blockIdx.y * 16;
    const int rowBase = blockIdx.x * GEMM_ROWS + wave * 64;

    // ---- stage W panel into LDS: wlds[kp*16+n] = {W[2kp][col0+n], W[2kp+1][col0+n]}
    const int npairs = (K >> 1) * 16;
    for (int idx = tid; idx < npairs; idx += GEMM_TB) {
        const int kp = idx >> 4, n = idx & 15;
        const float* wp = W + (size_t)(2 * kp) * Nc + col0 + n;
        float2 w2;
        w2.x = wp[0];
        w2.y = wp[Nc];
        wlds[idx] = w2;
    }
    __syncthreads();

    // ---- clamped per-lane 32-bit A element offsets (branch-free tails)
    uint32_t o0, o1, o2, o3;
    {
        int m0 = rowBase +  0 + l16; m0 = m0 < M ? m0 : M - 1;
        int m1 = rowBase + 16 + l16; m1 = m1 < M ? m1 : M - 1;
        int m2 = rowBase + 32 + l16; m2 = m2 < M ? m2 : M - 1;
        int m3 = rowBase + 48 + l16; m3 = m3 < M ? m3 : M - 1;
        o0 = (uint32_t)m0 * (uint32_t)K;
        o1 = (uint32_t)m1 * (uint32_t)K;
        o2 = (uint32_t)m2 * (uint32_t)K;
        o3 = (uint32_t)m3 * (uint32_t)K;
    }

    // ---- prologue: load k-step 0 fragments
    const uint32_t kk0 = 2u * half;
    float2 bw = wlds[half * 16 + l16];
    float2 t0 = *(const float2*)(A + o0 + kk0);
    float2 t1 = *(const float2*)(A + o1 + kk0);
    float2 t2 = *(const float2*)(A + o2 + kk0);
    float2 t3 = *(const float2*)(A + o3 + kk0);

    v8f acc0 = {}, acc1 = {}, acc2 = {}, acc3 = {};
    for (int k0 = 0; k0 < K; k0 += 4) {
        // prefetch next k-step (clamped on last iteration: dead reload)
        const int      kn  = (k0 + 4 < K) ? (k0 + 4) : k0;
        const uint32_t kkn = (uint32_t)kn + 2u * half;
        float2 bw_n = wlds[(kkn >> 1) * 16 + l16];
        float2 n0 = *(const float2*)(A + o0 + kkn);
        float2 n1 = *(const float2*)(A + o1 + kkn);
        float2 n2 = *(const float2*)(A + o2 + kkn);
        float2 n3 = *(const float2*)(A + o3 + kkn);

        // consume current fragments
        v2f b;  b.x  = bw.x; b.y  = bw.y;
        v2f a0; a0.x = t0.x; a0.y = t0.y;
        v2f a1; a1.x = t1.x; a1.y = t1.y;
        v2f a2; a2.x = t2.x; a2.y = t2.y;
        v2f a3; a3.x = t3.x; a3.y = t3.y;
        acc0 = __builtin_amdgcn_wmma_f32_16x16x4_f32(false, a0, false, b, (short)0, acc0, false, false);
        acc1 = __builtin_amdgcn_wmma_f32_16x16x4_f32(false, a1, false, b, (short)0, acc1, false, false);
        acc2 = __builtin_amdgcn_wmma_f32_16x16x4_f32(false, a2, false, b, (short)0, acc2, false, false);
        acc3 = __builtin_amdgcn_wmma_f32_16x16x4_f32(false, a3, false, b, (short)0, acc3, false, false);

        bw = bw_n; t0 = n0; t1 = n1; t2 = n2; t3 = n3;
    }

    const float bv = bias ? bias[col0 + l16] : 0.0f;
    const size_t colOff = (size_t)col0 + l16;
#pragma unroll
    for (int i = 0; i < 4; ++i) {
        const v8f acc = (i == 0) ? acc0 : (i == 1) ? acc1 : (i == 2) ? acc2 : acc3;
        const int rb = rowBase + i * 16 + 8 * half;
#pragma unroll
        for (int v = 0; v < 8; ++v) {
            const int r = rb + v;
            if (r < M) {
                float o = acc[v] + bv;
                if (relu) o = fmaxf(o, 0.0f);
                Out[(size_t)r * Nc + colOff] = o;
            }
        }
    }
}

// ---------------------------------------------------------------------------
// Attention helpers (C == 64 everywhere: lane handles a contiguous float2)
// ---------------------------------------------------------------------------
__device__ inline void atomicMaxF(float* addr, float v) {
    // monotone-in-float-order bit tricks: both paths only ever raise the value
    if (v >= 0.0f) atomicMax((int*)addr, __float_as_int(v));
    else           atomicMin((unsigned int*)addr, __float_as_uint(v));
}

__global__ void gtr_init_att(float* __restrict__ amax, float* __restrict__ denom,
                             float* __restrict__ numer, long nNH, long nNum)
{
    long t = (long)blockIdx.x * blockDim.x + threadIdx.x;
    if (t < nNum) numer[t] = 0.0f;
    if (t < nNH) { amax[t] = -INFINITY; denom[t] = 0.0f; }
}

// pass 1 (wave per (edge,head)): alpha = <q[dst], k[src]+eh[e]> / 8 ; segment max
__global__ void gtr_att_alpha(const float* __restrict__ q, const float* __restrict__ k,
                              const float* __restrict__ eh,
                              const int* __restrict__ src, const int* __restrict__ dst,
                              float* __restrict__ alpha, float* __restrict__ amax,
                              int E, int H)
{
    const int  C    = 64;
    long wid = ((long)blockIdx.x * blockDim.x + threadIdx.x) >> 5;
    const int lane = threadIdx.x & 31;
    if (wid >= (long)E * H) return;
    const int e = (int)(wid / H), h = (int)(wid % H);
    const int s = src[e], d = dst[e];
    const size_t off = (size_t)h * C + lane * 2;
    const float2 qv = *(const float2*)(q  + (size_t)d * H * C + off);
    const float2 kv = *(const float2*)(k  + (size_t)s * H * C + off);
    const float2 ev = *(const float2*)(eh + (size_t)e * H * C + off);
    float acc = qv.x * (kv.x + ev.x) + qv.y * (kv.y + ev.y);
#pragma unroll
    for (int o = 16; o; o >>= 1) acc += __shfl_xor(acc, o, 32);
    if (lane == 0) {
        acc *= 0.125f;                       // 1/sqrt(64)
        alpha[wid] = acc;
        atomicMaxF(amax + (size_t)d * H + h, acc);
    }
}

// pass 2 (wave per (edge,head)): ea=exp(alpha-amax); denom+=ea; numer[dst]+=ea*(v[src]+eh)
__global__ void gtr_att_scatter(const float* __restrict__ vmat, const float* __restrict__ eh,
                                const float* __restrict__ alpha, const float* __restrict__ amax,
                                const int* __restrict__ src, const int* __restrict__ dst,
                                float* __restrict__ denom, float* __restrict__ numer,
                                int E, int H)
{
    const int  C    = 64;
    long wid = ((long)blockIdx.x * blockDim.x + threadIdx.x) >> 5;
    const int lane = threadIdx.x & 31;
    if (wid >= (long)E * H) return;
    const int e = (int)(wid / H), h = (int)(wid % H);
    const int s = src[e], d = dst[e];
    const float ea = expf(alpha[wid] - amax[(size_t)d * H + h]);
    if (lane == 0) atomicAdd(denom + (size_t)d * H + h, ea);
    const size_t off = (size_t)h * C + lane * 2;
    const float2 vv = *(const float2*)(vmat + (size_t)s * H * C + off);
    const float2 ev = *(const float2*)(eh   + (size_t)e * H * C + off);
    float* __restrict__ np = numer + (size_t)d * H * C + off;
    atomicAdd(np + 0, ea * (vv.x + ev.x));
    atomicAdd(np + 1, ea * (vv.y + ev.y));
}

// pass 3: out = ELU(BN(numer/denom + skip))
__global__ void gtr_node_epilogue(const float* __restrict__ numer, const float* __restrict__ denom,
                                  const float* __restrict__ skip,
                                  const float* __restrict__ bg, const float* __restrict__ bb,
                                  const float* __restrict__ bm, const float* __restrict__ bv,
                                  float* __restrict__ out, int Nn, int HCl, int C)
{
    long t = (long)blockIdx.x * blockDim.x + threadIdx.x;
    if (t >= (long)Nn * HCl) return;
    int n  = (int)(t / HCl);
    int ch = (int)(t % HCl);
    int h  = ch / C;
    int H  = HCl / C;
    float den = denom[(size_t)n * H + h];
    float val = (den > 0.0f) ? numer[t] / den : 0.0f;   // nodes w/o in-edges -> 0
    val += skip[t];
    val = (val - bm[ch]) * (bg[ch] * rsqrtf(bv[ch] + 1e-5f)) + bb[ch];
    out[t] = (val > 0.0f) ? val : expm1f(val);          // ELU
}

// ---------------------------------------------------------------------------
// Pooling + head
// ---------------------------------------------------------------------------
__global__ void gtr_zero(float* __restrict__ p, long n)
{
    long t = (long)blockIdx.x * blockDim.x + threadIdx.x;
    if (t < n) p[t] = 0.0f;
}

__global__ void gtr_pool_sum(const float* __restrict__ h, const int* __restrict__ batch,
                             float* __restrict__ sums, float* __restrict__ cnt, int Nn)
{
    long t = (long)blockIdx.x * blockDim.x + threadIdx.x;
    if (t >= (long)Nn * NNODE_DIM) return;
    int n  = (int)(t / NNODE_DIM);
    int ch = (int)(t % NNODE_DIM);
    int g  = batch[n];
    atomicAdd(sums + (size_t)g * NNODE_DIM + ch, h[t]);
    if (ch == 0) atomicAdd(cnt + g, 1.0f);
}

__global__ void gtr_head_mlp(const float* __restrict__ sums, const float* __restrict__ cnt,
                             const float* __restrict__ w1, const float* __restrict__ b1,
                             const float* __restrict__ w2, const float* __restrict__ b2,
                             const float* __restrict__ w3, const float* __restrict__ b3,
                             float* __restrict__ out, int Gn)
{
    int g = blockIdx.x * blockDim.x + threadIdx.x;
    if (g >= Gn) return;
    float c = fmaxf(cnt[g], 1.0f);
    float gv[64];
#pragma unroll
    for (int i = 0; i < 64; ++i) gv[i] = sums[(size_t)g * 64 + i] / c;
    float r1[32];
    for (int j = 0; j < 32; ++j) {
        float s = b1[j];
        for (int i = 0; i < 64; ++i) s += gv[i] * w1[i * 32 + j];
        r1[j] = fmaxf(s, 0.0f);
    }
    float r2[16];
    for (int j = 0; j < 16; ++j) {
        float s = b2[j];
        for (int i = 0; i < 32; ++i) s += r1[i] * w2[i * 16 + j];
        r2[j] = fmaxf(s, 0.0f);
    }
    float o = b3[0];
    for (int i = 0; i < 16; ++i) o += r2[i] * w3[i];
    out[g] = o;
}

// ---------------------------------------------------------------------------
// Orchestration
// ---------------------------------------------------------------------------
extern "C" void kernel_launch(void* const* d_in, const int* in_sizes, int n_in,
                              void* d_out, int out_size, void* d_ws, size_t ws_size,
                              hipStream_t stream)
{
    const float* x         = (const float*)d_in[0];
    const int*   edge_idx  = (const int*)  d_in[1];
    const float* edge_attr = (const float*)d_in[2];
    const int*   batch     = (const int*)  d_in[3];
    const float* ee_w1 = (const float*)d_in[4];  const float* ee_b1 = (const float*)d_in[5];
    const float* ee_w2 = (const float*)d_in[6];  const float* ee_b2 = (const float*)d_in[7];
    // conv params: wq,bq,wk,bk,wv,bv,we,ws,bs
    const float* c1[9]; for (int i = 0; i < 9; ++i) c1[i] = (const float*)d_in[8  + i];
    const float* c2[9]; for (int i = 0; i < 9; ++i) c2[i] = (const float*)d_in[17 + i];
    const float* c3[9]; for (int i = 0; i < 9; ++i) c3[i] = (const float*)d_in[26 + i];
    const float* bn1[4]; for (int i = 0; i < 4; ++i) bn1[i] = (const float*)d_in[35 + i];
    const float* bn2[4]; for (int i = 0; i < 4; ++i) bn2[i] = (const float*)d_in[39 + i];
    const float* bn3[4]; for (int i = 0; i < 4; ++i) bn3[i] = (const float*)d_in[43 + i];
    const float* r_w1 = (const float*)d_in[47]; const float* r_b1 = (const float*)d_in[48];
    const float* r_w2 = (const float*)d_in[49]; const float* r_b2 = (const float*)d_in[50];
    const float* r_w3 = (const float*)d_in[51]; const float* r_b3 = (const float*)d_in[52];

    const int N = in_sizes[0] / 64;     // 50000
    const int E = in_sizes[1] / 2;      // 400000
    const int G = out_size;             // 128
    const int* src = edge_idx;          // edge_index[0]
    const int* dst = edge_idx + E;      // edge_index[1]

    // workspace carve-out (fp32)
    float* p = (float*)d_ws;
    float* e_buf = p; p += (size_t)E * 64;    // edge embeddings (shared across layers)
    float* eh    = p; p += (size_t)E * 256;   // per-layer per-head edge emb (encoder tmp aliases)
    float* qb    = p; p += (size_t)N * 256;
    float* kb    = p; p += (size_t)N * 256;
    float* vb    = p; p += (size_t)N * 256;
    float* skip  = p; p += (size_t)N * 256;
    float* numer = p; p += (size_t)N * 256;
    float* hA    = p; p += (size_t)N * 256;
    float* hB    = p; p += (size_t)N * 256;
    float* alpha = p; p += (size_t)E * 4;
    float* amax  = p; p += (size_t)N * 4;
    float* denom = p; p += (size_t)N * 4;
    float* sums  = p; p += (size_t)G * 64;
    float* cnt   = p; p += (size_t)G;

    const int TB = 256;
    auto blocks = [](long n, int tb) { return (unsigned)((n + tb - 1) / tb); };
    auto ggrid  = [](int M, int nc) { return dim3((M + GEMM_ROWS - 1) / GEMM_ROWS, nc / 16); };

    // ---- edge encoder: e = relu(ea @ w1 + b1) @ w2 + b2
    float* etmp = eh;  // alias: used before eh is needed
    gtr_wmma_gemm<<<ggrid(E, 64), GEMM_TB, 0, stream>>>(edge_attr, ee_w1, ee_b1, etmp, E, 16, 64, 1);
    gtr_wmma_gemm<<<ggrid(E, 64), GEMM_TB, 0, stream>>>(etmp, ee_w2, ee_b2, e_buf, E, 64, 64, 0);

    // ---- one TransformerConv layer
    auto run_layer = [&](const float* xin, int din, int H, int C,
                         const float* const cp[9], const float* const bn[4], float* hout) {
        const int HCl = H * C;
        gtr_wmma_gemm<<<ggrid(N, HCl), GEMM_TB, 0, stream>>>(xin, cp[0], cp[1], qb, N, din, HCl, 0);   // q
        gtr_wmma_gemm<<<ggrid(N, HCl), GEMM_TB, 0, stream>>>(xin, cp[2], cp[3], kb, N, din, HCl, 0);   // k
        gtr_wmma_gemm<<<ggrid(N, HCl), GEMM_TB, 0, stream>>>(xin, cp[4], cp[5], vb, N, din, HCl, 0);   // v
        gtr_wmma_gemm<<<ggrid(E, HCl), GEMM_TB, 0, stream>>>(e_buf, cp[6], nullptr, eh, E, 64, HCl, 0); // eh
        gtr_wmma_gemm<<<ggrid(N, HCl), GEMM_TB, 0, stream>>>(xin, cp[7], cp[8], skip, N, din, HCl, 0); // skip
        gtr_init_att<<<blocks((long)N * HCl, TB), TB, 0, stream>>>(amax, denom, numer,
                                                                   (long)N * H, (long)N * HCl);
        gtr_att_alpha<<<blocks((long)E * H * 32, TB), TB, 0, stream>>>(qb, kb, eh, src, dst,
                                                                       alpha, amax, E, H);
        gtr_att_scatter<<<blocks((long)E * H * 32, TB), TB, 0, stream>>>(vb, eh, alpha, amax, src, dst,
                                                                         denom, numer, E, H);
        gtr_node_epilogue<<<blocks((long)N * HCl, TB), TB, 0, stream>>>(numer, denom, skip,
                                                                        bn[0], bn[1], bn[2], bn[3],
                                                                        hout, N, HCl, C);
    };

    run_layer(x,  64,  4, 64, c1, bn1, hA);    // conv1: 64  -> 256
    run_layer(hA, 256, 4, 64, c2, bn2, hB);    // conv2: 256 -> 256
    run_layer(hB, 256, 1, 64, c3, bn3, hA);    // conv3: 256 -> 64 (heads=1)

    // ---- global mean pool + MLP head
    gtr_zero<<<blocks((long)G * 64 + G, TB), TB, 0, stream>>>(sums, (long)G * 64 + (long)G);
    gtr_pool_sum<<<blocks((long)N * 64, TB), TB, 0, stream>>>(hA, batch, sums, cnt, N);
    gtr_head_mlp<<<(G + 63) / 64, 64, 0, stream>>>(sums, cnt, r_w1, r_b1, r_w2, r_b2, r_w3, r_b3,
                                                   (float*)d_out, G);
}